// Camera_back_projection_layer_68762426409753
// MI455X (gfx1250) — compile-verified
//
#include <hip/hip_runtime.h>
#include <hip/hip_bf16.h>

// Camera back-projection to truncated-distance voxel grid.
// out[n,x,y,z] = in_frustum ? max(0, 1 - 128*|zc - depth[n,vi,ui]|) : 0
//
// Store-bandwidth bound: 134 MB output @ 23.3 TB/s => ~5.8us floor.
// Strategy: precompute the (identical) u/v projection tables once (262K IEEE
// divisions instead of 2 per voxel), then a pure gather+store main kernel with
// 128-bit coalesced stores along z and global_prefetch_b8 pipelining of the
// scattered depth gathers (depth working set: 14.7 MB, L2-resident).

namespace {
constexpr int   RESV   = 128;
constexpr int   IMG    = 480;
constexpr int   NB     = 16;
constexpr float INV_RES = 1.0f / 128.0f;
constexpr float HALF_IMG = (IMG - 1) * 0.5f;   // 239.5
}

// ---------------------------------------------------------------------------
// Stage 1: tab[n][c][z] = clamp(round(fl*coord/zc + 239.5), 0, 479)
//                         | (valid << 31),  valid = (u in [0,479]) && zc>0
// Same table serves both u (coord=x) and v (coord=y) since H==W and the
// coordinate grids are identical.
// ---------------------------------------------------------------------------
__global__ __launch_bounds__(256) void cbp_build_table(
    const float* __restrict__ fl, const float* __restrict__ cd,
    int* __restrict__ tab) {
  int idx = blockIdx.x * blockDim.x + threadIdx.x;   // n*16384 + c*128 + z
  int z = idx & (RESV - 1);
  int c = (idx >> 7) & (RESV - 1);
  int n = idx >> 14;

  float coord  = ((float)c + 0.5f) * INV_RES - 0.5f;     // exact
  float zcoord = ((float)z + 0.5f) * INV_RES - 0.5f;     // exact
  float zc = cd[n] - zcoord;                              // one RNE rounding
  float u  = (fl[n] * coord) / zc + HALF_IMG;             // mul, IEEE div, add
  int ui = (int)rintf(u);                                 // round-half-even
  ui = ui < 0 ? 0 : (ui > IMG - 1 ? IMG - 1 : ui);
  bool valid = (u >= 0.0f) && (u <= (float)(IMG - 1)) && (zc > 0.0f);
  tab[idx] = ui | (valid ? (int)0x80000000 : 0);
}

// ---------------------------------------------------------------------------
// Stage 2: one block per (n, x). threadIdx.x (32 lanes) covers z in quads of 4
// (int4/float4 128-bit traffic), threadIdx.y (8) strides the y loop.
// v-table loads are register double-buffered; next iteration's 4 depth-gather
// addresses are warmed with global_prefetch_b8 (gfx1250) while computing.
// ---------------------------------------------------------------------------
__global__ __launch_bounds__(256) void cbp_main(
    const float* __restrict__ depth, const float* __restrict__ cd,
    const int* __restrict__ tab, float* __restrict__ out) {
  const int n  = blockIdx.x >> 7;
  const int x  = blockIdx.x & (RESV - 1);
  const int z0 = (int)threadIdx.x << 2;     // 0,4,...,124
  const int ty = (int)threadIdx.y;          // 0..7

  const int* tn = tab + (n << 14);          // this batch's table [c][z]
  const int4 up4 = *(const int4*)(tn + (x << 7) + z0);
  const int up[4] = {up4.x, up4.y, up4.z, up4.w};
  int ui[4];
#pragma unroll
  for (int j = 0; j < 4; ++j) ui[j] = up[j] & 0xFFFF;

  const float cdn = cd[n];
  float zc[4];
#pragma unroll
  for (int j = 0; j < 4; ++j)
    zc[j] = cdn - (((float)(z0 + j) + 0.5f) * INV_RES - 0.5f);

  const float* __restrict__ dn = depth + (size_t)n * (IMG * IMG);
  float* __restrict__ on =
      out + (((size_t)n * RESV + x) * RESV) * RESV + z0;

  int4 vp = *(const int4*)(tn + (ty << 7) + z0);

#pragma unroll 4
  for (int y = ty; y < RESV; y += 8) {
    const int yn = y + 8;
    int4 vpn = vp;
    if (yn < RESV) {
      vpn = *(const int4*)(tn + (yn << 7) + z0);
      const int pn[4] = {vpn.x, vpn.y, vpn.z, vpn.w};
#pragma unroll
      for (int j = 0; j < 4; ++j)   // gfx1250 global_prefetch_b8
        __builtin_prefetch(dn + (pn[j] & 0xFFFF) * IMG + ui[j]);
    }

    const int pv[4] = {vp.x, vp.y, vp.z, vp.w};
    float rr[4];
#pragma unroll
    for (int j = 0; j < 4; ++j) {
      const int vi = pv[j] & 0xFFFF;
      const float d = dn[vi * IMG + ui[j]];
      const bool ok = ((pv[j] & up[j]) < 0) && (d > 0.0f);
      const float a = fabsf(zc[j] - d);
      // match reference's separate mul/sub (no FMA contraction)
      float t = __fsub_rn(1.0f, __fmul_rn(128.0f, a));
      t = fmaxf(t, 0.0f);           // a >= 1/128 (truncation) -> exactly 0
      rr[j] = ok ? t : 0.0f;        // out-of-frustum -> 1 - 128*(1/128) = 0
    }
    *(float4*)(on + (size_t)y * RESV) = make_float4(rr[0], rr[1], rr[2], rr[3]);
    vp = vpn;
  }
}

// ---------------------------------------------------------------------------
// Fallback (no workspace): fused kernel, u-divisions hoisted (4/thread),
// v-division per (y, z-lane) inside the loop. Correct, just more VALU.
// ---------------------------------------------------------------------------
__global__ __launch_bounds__(256) void cbp_fused(
    const float* __restrict__ depth, const float* __restrict__ fl,
    const float* __restrict__ cd, float* __restrict__ out) {
  const int n  = blockIdx.x >> 7;
  const int x  = blockIdx.x & (RESV - 1);
  const int z0 = (int)threadIdx.x << 2;
  const int ty = (int)threadIdx.y;

  const float f = fl[n], cdn = cd[n];
  const float xc = ((float)x + 0.5f) * INV_RES - 0.5f;

  float zc[4]; int ui[4]; bool uok[4];
#pragma unroll
  for (int j = 0; j < 4; ++j) {
    zc[j] = cdn - (((float)(z0 + j) + 0.5f) * INV_RES - 0.5f);
    float u = (f * xc) / zc[j] + HALF_IMG;
    int uu = (int)rintf(u);
    ui[j] = uu < 0 ? 0 : (uu > IMG - 1 ? IMG - 1 : uu);
    uok[j] = (u >= 0.0f) && (u <= (float)(IMG - 1)) && (zc[j] > 0.0f);
  }

  const float* __restrict__ dn = depth + (size_t)n * (IMG * IMG);
  float* __restrict__ on =
      out + (((size_t)n * RESV + x) * RESV) * RESV + z0;

  for (int y = ty; y < RESV; y += 8) {
    const float yc = ((float)y + 0.5f) * INV_RES - 0.5f;
    float rr[4];
#pragma unroll
    for (int j = 0; j < 4; ++j) {
      float v = (f * yc) / zc[j] + HALF_IMG;
      int vv = (int)rintf(v);
      vv = vv < 0 ? 0 : (vv > IMG - 1 ? IMG - 1 : vv);
      const float d = dn[vv * IMG + ui[j]];
      const bool ok =
          uok[j] && (v >= 0.0f) && (v <= (float)(IMG - 1)) && (d > 0.0f);
      const float a = fabsf(zc[j] - d);
      float t = __fsub_rn(1.0f, __fmul_rn(128.0f, a));
      t = fmaxf(t, 0.0f);
      rr[j] = ok ? t : 0.0f;
    }
    *(float4*)(on + (size_t)y * RESV) = make_float4(rr[0], rr[1], rr[2], rr[3]);
  }
}

extern "C" void kernel_launch(void* const* d_in, const int* in_sizes, int n_in,
                              void* d_out, int out_size, void* d_ws,
                              size_t ws_size, hipStream_t stream) {
  const float* depth = (const float*)d_in[0];   // [16,1,480,480] f32
  const float* fl    = (const float*)d_in[1];   // [16,1] f32
  const float* cd    = (const float*)d_in[2];   // [16,1] f32
  float* out = (float*)d_out;                   // [16,1,128,128,128] f32

  const size_t tab_bytes = (size_t)NB * RESV * RESV * sizeof(int);  // 1 MB
  if (d_ws != nullptr && ws_size >= tab_bytes) {
    int* tab = (int*)d_ws;
    cbp_build_table<<<(NB * RESV * RESV) / 256, 256, 0, stream>>>(fl, cd, tab);
    cbp_main<<<NB * RESV, dim3(32, 8), 0, stream>>>(depth, cd, tab, out);
  } else {
    cbp_fused<<<NB * RESV, dim3(32, 8), 0, stream>>>(depth, fl, cd, out);
  }
}